// GCN_90778428768456
// MI455X (gfx1250) — compile-verified
//
#include <hip/hip_runtime.h>
#include <hip/hip_bf16.h>

#define FEAT 64

typedef __attribute__((ext_vector_type(2))) float v2f;
typedef __attribute__((ext_vector_type(8))) float v8f;

// ---------------------------------------------------------------------------
// Degree / normalization (computed once, reused by all 3 layers)
// ---------------------------------------------------------------------------
__global__ void zero_f32_kernel(float* __restrict__ p, int n) {
    int i = blockIdx.x * blockDim.x + threadIdx.x;
    if (i < n) p[i] = 0.0f;
}

__global__ void degree_kernel(const int* __restrict__ dst, float* __restrict__ deg,
                              int n_edges) {
    int e = blockIdx.x * blockDim.x + threadIdx.x;
    if (e < n_edges) atomicAdd(&deg[dst[e]], 1.0f);
}

__global__ void dinv_kernel(const float* __restrict__ deg, float* __restrict__ dinv,
                            int n) {
    int i = blockIdx.x * blockDim.x + threadIdx.x;
    if (i < n) dinv[i] = rsqrtf(deg[i] + 1.0f);   // +1 self-loop; deg >= 1
}

// ---------------------------------------------------------------------------
// Dense transform: H = (RELU?)(X) @ W   via V_WMMA_F32_16X16X4_F32
// One wave32 handles a 16-row tile across all 64 output columns.
//
// W is staged into LDS PRE-PAIRED along K:  sW[t][n] = {W[2t][n], W[2t+1][n]}
// so every B fragment is one aligned ds_load_b64 straight into the even VGPR
// pair the WMMA consumes — no register repacking between DS loads and WMMAs.
//
// A frag (16x4 f32, 2 VGPRs): lane l: M = l&15, K = {ka, ka+1},
//   ka = k + 2*(l>>4)  (lanes 0-15 -> K k,k+1 ; lanes 16-31 -> K k+2,k+3)
// B frag (4x16 f32, 2 VGPRs): lane l: N = l&15, same K split -> pair index
//   t = k/2 + (l>>4).
// C/D (16x16 f32, 8 VGPRs): element v of lane l -> M = v + 8*(l>>4), N = l&15.
// ---------------------------------------------------------------------------
template <bool RELU>
__global__ void __launch_bounds__(256)
gcn_gemm_wmma(const float* __restrict__ X, const float* __restrict__ W,
              float* __restrict__ H, int n_nodes) {
    __shared__ v2f sW[(FEAT / 2) * FEAT];          // [t][n] pairs, 16KB
    const int tid = threadIdx.x;
    for (int i = tid; i < FEAT * FEAT; i += 256) {
        const int r = i >> 6;                      // K row
        const int c = i & 63;                      // N col
        // component (r&1) of pair (r>>1) at column c
        ((float*)sW)[(((r >> 1) * FEAT + c) << 1) + (r & 1)] = W[i];
    }
    __syncthreads();

    const int lane    = tid & 31;
    const int wave    = tid >> 5;
    const int n_tiles = (n_nodes + 15) >> 4;
    const int tile    = blockIdx.x * 8 + wave;     // 8 waves per block
    if (tile >= n_tiles) return;                   // wave-uniform exit

    const int r0    = tile << 4;
    const int m     = lane & 15;
    const int thalf = lane >> 4;                   // 0 or 1 (K-pair select)

    int row = r0 + m;
    if (row >= n_nodes) row = n_nodes - 1;         // clamp loads; stores masked
    const float* __restrict__ xrow = X + (size_t)row * FEAT;

    v8f acc0 = {}, acc1 = {}, acc2 = {}, acc3 = {};

#pragma unroll
    for (int k = 0; k < FEAT; k += 4) {
        const int ka = k + (thalf << 1);
        v2f a;
        a.x = xrow[ka];
        a.y = xrow[ka + 1];
        if (RELU) {
            a.x = fmaxf(a.x, 0.0f);
            a.y = fmaxf(a.y, 0.0f);
        }
        const v2f* __restrict__ wp = sW + ((k >> 1) + thalf) * FEAT + m;
        const v2f b0 = wp[0];                      // single ds_load_b64 each
        const v2f b1 = wp[16];
        const v2f b2 = wp[32];
        const v2f b3 = wp[48];

        acc0 = __builtin_amdgcn_wmma_f32_16x16x4_f32(false, a, false, b0, (short)0, acc0, false, false);
        acc1 = __builtin_amdgcn_wmma_f32_16x16x4_f32(false, a, false, b1, (short)0, acc1, false, false);
        acc2 = __builtin_amdgcn_wmma_f32_16x16x4_f32(false, a, false, b2, (short)0, acc2, false, false);
        acc3 = __builtin_amdgcn_wmma_f32_16x16x4_f32(false, a, false, b3, (short)0, acc3, false, false);
    }

    const int mhi = thalf << 3;                    // 0 or 8
#pragma unroll
    for (int v = 0; v < 8; ++v) {
        const int grow = r0 + v + mhi;
        if (grow < n_nodes) {
            float* __restrict__ hrow = H + (size_t)grow * FEAT;
            hrow[ 0 + m] = acc0[v];
            hrow[16 + m] = acc1[v];
            hrow[32 + m] = acc2[v];
            hrow[48 + m] = acc3[v];
        }
    }
}

// ---------------------------------------------------------------------------
// out = h * dinv^2 (self-loop term) + bias   — writes EVERY output element
// (initializes the accumulation buffer; float4 vectorized)
// ---------------------------------------------------------------------------
__global__ void selfloop_bias_kernel(const float4* __restrict__ H,
                                     const float* __restrict__ dinv,
                                     const float* __restrict__ bias,
                                     float4* __restrict__ out, int total4) {
    int i = blockIdx.x * blockDim.x + threadIdx.x;
    if (i >= total4) return;
    const int node = i >> 4;              // FEAT/4 = 16 float4 per node
    const int f    = (i & 15) << 2;
    const float di = dinv[node];
    const float w  = di * di;
    const float4 h = H[i];
    const float4 b = *(const float4*)(bias + f);
    float4 o;
    o.x = h.x * w + b.x;
    o.y = h.y * w + b.y;
    o.z = h.z * w + b.z;
    o.w = h.w * w + b.w;
    out[i] = o;
}

// ---------------------------------------------------------------------------
// Edge aggregation: out[dst] += h[src] * dinv[src]*dinv[dst]
//
// Block stages 64 edges' metadata (src, dst, precomputed weight) into LDS
// with coalesced loads (one thread per edge), then each of the 8 waves walks
// 8 edges: metadata comes back via conflict-free LDS broadcast (uniform
// address), each lane gathers 2 features (float2) and issues 2 f32 atomics.
// Index/dinv traffic is deduplicated ~32x vs the naive wave-per-edge scheme;
// remaining traffic is the L2-resident 256B gather + 256B of atomics per
// edge, which is the scatter's roofline floor.
// ---------------------------------------------------------------------------
#define EDGES_PER_BLOCK 64

__global__ void __launch_bounds__(256)
edge_scatter_kernel(const float* __restrict__ H,
                    const int* __restrict__ src,
                    const int* __restrict__ dst,
                    const float* __restrict__ dinv,
                    float* __restrict__ out, int n_edges) {
    __shared__ int   s_src[EDGES_PER_BLOCK];
    __shared__ int   s_dst[EDGES_PER_BLOCK];
    __shared__ float s_w[EDGES_PER_BLOCK];

    const int tid = threadIdx.x;
    const int e0  = blockIdx.x * EDGES_PER_BLOCK;

    if (tid < EDGES_PER_BLOCK) {
        const int e = e0 + tid;
        int s = 0, d = 0;
        float w = 0.0f;
        if (e < n_edges) {
            s = src[e];
            d = dst[e];
            w = dinv[s] * dinv[d];
        }
        s_src[tid] = s;
        s_dst[tid] = d;
        s_w[tid]   = w;
    }
    __syncthreads();

    const int wave = tid >> 5;
    const int lane = tid & 31;
    const int f    = lane << 1;

#pragma unroll
    for (int i = 0; i < EDGES_PER_BLOCK / 8; ++i) {
        const int le = wave * (EDGES_PER_BLOCK / 8) + i;
        const int e  = e0 + le;
        if (e >= n_edges) break;           // wave-uniform
        const int   s = s_src[le];         // LDS broadcast reads
        const int   d = s_dst[le];
        const float w = s_w[le];
        const float2 h = *(const float2*)(H + (size_t)s * FEAT + f);
        float* op = out + (size_t)d * FEAT + f;
        atomicAdd(op + 0, h.x * w);
        atomicAdd(op + 1, h.y * w);
    }
}

// ---------------------------------------------------------------------------
extern "C" void kernel_launch(void* const* d_in, const int* in_sizes, int n_in,
                              void* d_out, int out_size, void* d_ws, size_t ws_size,
                              hipStream_t stream) {
    const float* x  = (const float*)d_in[0];
    const int*   ei = (const int*)d_in[1];
    const float* W1 = (const float*)d_in[2];
    const float* b1 = (const float*)d_in[3];
    const float* W2 = (const float*)d_in[4];
    const float* b2 = (const float*)d_in[5];
    const float* W3 = (const float*)d_in[6];
    const float* b3 = (const float*)d_in[7];
    float* out = (float*)d_out;

    const int n_nodes = in_sizes[0] / FEAT;
    const int n_edges = in_sizes[1] / 2;
    const int* src = ei;
    const int* dst = ei + n_edges;

    // workspace layout (floats): hbuf | abuf | deg | dinv
    float* ws   = (float*)d_ws;
    float* hbuf = ws;
    float* abuf = hbuf + (size_t)n_nodes * FEAT;
    float* deg  = abuf + (size_t)n_nodes * FEAT;
    float* dinv = deg + n_nodes;

    const int total  = n_nodes * FEAT;
    const int total4 = total / 4;

    const int node_blocks = (n_nodes + 255) / 256;
    const int edge_blocks = (n_edges + 255) / 256;
    const int n_tiles     = (n_nodes + 15) / 16;
    const int gemm_blocks = (n_tiles + 7) / 8;                 // 8 waves/block
    const int init_blocks = (total4 + 255) / 256;
    const int scat_blocks = (n_edges + EDGES_PER_BLOCK - 1) / EDGES_PER_BLOCK;

    // --- normalization (once) ---
    zero_f32_kernel<<<node_blocks, 256, 0, stream>>>(deg, n_nodes);
    degree_kernel<<<edge_blocks, 256, 0, stream>>>(dst, deg, n_edges);
    dinv_kernel<<<node_blocks, 256, 0, stream>>>(deg, dinv, n_nodes);

    // --- layer 1: h1 = x @ W1 ; out1 = agg + h1*dinv^2 + b1 (-> abuf) ---
    gcn_gemm_wmma<false><<<gemm_blocks, 256, 0, stream>>>(x, W1, hbuf, n_nodes);
    selfloop_bias_kernel<<<init_blocks, 256, 0, stream>>>(
        (const float4*)hbuf, dinv, b1, (float4*)abuf, total4);
    edge_scatter_kernel<<<scat_blocks, 256, 0, stream>>>(
        hbuf, src, dst, dinv, abuf, n_edges);

    // --- layer 2: h2 = relu(out1) @ W2 ; out2 -> abuf (overwrite ok) ---
    gcn_gemm_wmma<true><<<gemm_blocks, 256, 0, stream>>>(abuf, W2, hbuf, n_nodes);
    selfloop_bias_kernel<<<init_blocks, 256, 0, stream>>>(
        (const float4*)hbuf, dinv, b2, (float4*)abuf, total4);
    edge_scatter_kernel<<<scat_blocks, 256, 0, stream>>>(
        hbuf, src, dst, dinv, abuf, n_edges);

    // --- layer 3: h3 = relu(out2) @ W3 ; final -> d_out (no relu) ---
    gcn_gemm_wmma<true><<<gemm_blocks, 256, 0, stream>>>(abuf, W3, hbuf, n_nodes);
    selfloop_bias_kernel<<<init_blocks, 256, 0, stream>>>(
        (const float4*)hbuf, dinv, b3, (float4*)out, total4);
    edge_scatter_kernel<<<scat_blocks, 256, 0, stream>>>(
        hbuf, src, dst, dinv, out, n_edges);
}